// FillNet_87643102642307
// MI455X (gfx1250) — compile-verified
//
#include <hip/hip_runtime.h>

// ---------------------------------------------------------------------------
// RBF kernel regression:  out[n] = (sum_m W2[m]*K[n,m]) / (sum_m K[n,m]),
//   K[n,m] = exp(-||X[n]-P[m]||^2 / (2*sigma^2)),  N=8192, M=16384, dim=2.
//
// MI455X reasoning: inputs ~256KB -> HBM time negligible at 23.3 TB/s; the
// cost is 134M exps + accumulate VALU. Squared distance as a rank-4 GEMM,
// with the exp2 scale constant folded into the B operand:
//   [x0,x1,|x|^2,1] . C2*[-2p0,-2p1,1,|p|^2]  ->  V_WMMA_F32_16X16X4_F32
// one WMMA = a full 16x16 tile of ready-to-exp2 f32 arguments. Inner loop:
//   ds_load_b96 -> v_wmma -> 8x v_exp_f32 + 4x v_pk_fma + 4x v_pk_add,
// branch-free; the v_exp_f32 trans rate is the throughput floor and all
// VALU work co-executes under it.
// ---------------------------------------------------------------------------

typedef __attribute__((ext_vector_type(2))) float v2f;
typedef __attribute__((ext_vector_type(8))) float v8f;

#define TILE_M          16      // m-columns per WMMA tile
#define LDS_M           1024    // centers staged per LDS refill (2*16KB = 32KB)
#define WAVES_PER_BLOCK 8
#define ROWS_PER_WAVE   16
#define ROWS_PER_BLOCK  (WAVES_PER_BLOCK * ROWS_PER_WAVE)   // 128
#define BLOCK_THREADS   (WAVES_PER_BLOCK * 32)              // 256 (8 waves, wave32)

// exp(-d / (2*sigma^2)) == exp2(d * C2),  sigma = 2.2
#define SIGMA_SQ 4.84f
#define C2 (-1.0f / (2.0f * SIGMA_SQ * 0.69314718055994531f))

__global__ __launch_bounds__(BLOCK_THREADS)
void rbf_pass1(const float* __restrict__ X,
               const float* __restrict__ P,
               const float* __restrict__ W2,
               float* __restrict__ partial,   // [num: MC*N][den: MC*N]
               int N, int M, int MC)
{
    // Pre-split, pre-scaled B operands per half-wave (ISA 7.12.2 B layout):
    //   shB[j]         = (C2*-2p0, C2*-2p1, w2, 0)  -> rows K=0,1 (lanes 0-15)
    //   shB[LDS_M + j] = (C2,      C2*|p|^2, w2, 0) -> rows K=2,3 (lanes 16-31)
    // so the WMMA result is directly the exp2 argument C2*||x-p||^2.
    __shared__ float4 shB[2 * LDS_M];

    const int lane = threadIdx.x & 31;
    const int wave = threadIdx.x >> 5;
    const int col  = lane & 15;                  // m-column within half-wave
    const int n0   = blockIdx.x * ROWS_PER_BLOCK + wave * ROWS_PER_WAVE;

    // Branch-free per-lane LDS base select (single v_cndmask at init).
    const int bofs = (lane < 16) ? 0 : LDS_M;

    const int chunk  = M / MC;
    const int mStart = blockIdx.y * chunk;
    const int mEnd   = mStart + chunk;

    // --- A matrix (16x4 f32): row n = [x0, x1, |x|^2, 1]
    // Lanes 0-15 hold K=0,1 in v0,v1 ; lanes 16-31 hold K=2,3.
    const float2 x  = ((const float2*)X)[n0 + col];
    const float  xs = x.x * x.x + x.y * x.y;
    v2f a;
    a.x = (lane < 16) ? x.x : xs;
    a.y = (lane < 16) ? x.y : 1.0f;

    v8f num = {0.f,0.f,0.f,0.f,0.f,0.f,0.f,0.f};
    v8f den = {0.f,0.f,0.f,0.f,0.f,0.f,0.f,0.f};

    for (int base = mStart; base < mEnd; base += LDS_M) {
        const int cnt = min(LDS_M, mEnd - base);

        // Cooperative stage: precompute both half-wave B quads per center,
        // with the exp2 constant folded in.
        for (int j = threadIdx.x; j < cnt; j += BLOCK_THREADS) {
            const int    m = base + j;
            const float2 p = ((const float2*)P)[m];
            const float  w = W2[m];
            shB[j]         = make_float4(C2 * -2.0f * p.x, C2 * -2.0f * p.y, w, 0.f);
            shB[LDS_M + j] = make_float4(C2, C2 * (p.x * p.x + p.y * p.y),   w, 0.f);
        }
        __syncthreads();

#pragma unroll 4
        for (int t = 0; t < cnt; t += TILE_M) {
            const float4 q = shB[bofs + t + col];   // one ds_load_b96

            v2f b; b.x = q.x; b.y = q.y;            // lane's 2 B-rows
            const float w2v = q.z;                  // W2 for this column m

            v8f c = {0.f,0.f,0.f,0.f,0.f,0.f,0.f,0.f};
            // D[n][m] = C2 * ||X[n]-P[m]||^2  (exact f32, ready for exp2)
            v8f d = __builtin_amdgcn_wmma_f32_16x16x4_f32(
                false, a, false, b, (short)0, c, false, false);

#pragma unroll
            for (int i = 0; i < 8; ++i) {
                const float k = __builtin_amdgcn_exp2f(d[i]);
                num[i] += k * w2v;
                den[i] += k;
            }
        }
        __syncthreads();
    }

    // Each 16-lane half-wave holds partials for 8 rows, spread across 16
    // m-columns: butterfly-reduce within groups of 16 lanes.
#pragma unroll
    for (int i = 0; i < 8; ++i) {
        float nv = num[i], dv = den[i];
        for (int off = 1; off < 16; off <<= 1) {
            nv += __shfl_xor(nv, off, 16);
            dv += __shfl_xor(dv, off, 16);
        }
        num[i] = nv; den[i] = dv;
    }

    if (col == 0) {  // lane 0 -> rows n0..n0+7 ; lane 16 -> rows n0+8..n0+15
        const int rowBase = n0 + ((lane >> 4) << 3);
#pragma unroll
        for (int i = 0; i < 8; ++i) {
            const size_t n = (size_t)(rowBase + i);
            partial[(size_t)blockIdx.y * N + n]                  = num[i];
            partial[(size_t)MC * N + (size_t)blockIdx.y * N + n] = den[i];
        }
    }
}

__global__ __launch_bounds__(256)
void rbf_pass2(const float* __restrict__ partial,
               float* __restrict__ out, int N, int MC)
{
    const int n = blockIdx.x * blockDim.x + threadIdx.x;
    if (n >= N) return;
    float num = 0.f, den = 0.f;
    for (int c = 0; c < MC; ++c) {
        num += partial[(size_t)c * N + n];
        den += partial[(size_t)MC * N + (size_t)c * N + n];
    }
    out[n] = num / den;
}

extern "C" void kernel_launch(void* const* d_in, const int* in_sizes, int n_in,
                              void* d_out, int out_size, void* d_ws, size_t ws_size,
                              hipStream_t stream)
{
    const float* X  = (const float*)d_in[0];
    const float* P  = (const float*)d_in[1];
    const float* W2 = (const float*)d_in[2];
    float* out      = (float*)d_out;

    const int N = in_sizes[0] / 2;   // 8192
    const int M = in_sizes[2];       // 16384 (W2 length)

    // Split M across grid.y for occupancy; partials go to d_ws (no float
    // atomics -> bitwise deterministic). Shrink MC if d_ws is small.
    int MC = 8;
    while (MC > 1 &&
           ((size_t)MC * (size_t)N * 2u * sizeof(float) > ws_size ||
            (M % (MC * TILE_M)) != 0))
        MC >>= 1;

    float* partial = (float*)d_ws;

    dim3 grid1(N / ROWS_PER_BLOCK, MC);
    rbf_pass1<<<grid1, BLOCK_THREADS, 0, stream>>>(X, P, W2, partial, N, M, MC);

    dim3 grid2((N + 255) / 256);
    rbf_pass2<<<grid2, 256, 0, stream>>>(partial, out, N, MC);
}